// BilateralFilter_5772436046180
// MI455X (gfx1250) — compile-verified
//
#include <hip/hip_runtime.h>
#include <hip/hip_bf16.h>

// Bilateral 5x5 filter, f32, reflect padding.
// B*C = 12 planes of 512x512. LDS-tiled stencil with CDNA5 async
// global->LDS DMA staging (ASYNCcnt) and hardware exp2/rcp weight math.

#define KS      5
#define PADR    2
#define IMG_H   512
#define IMG_W   512
#define TILE_W  64
#define TILE_H  64
#define LDS_W   (TILE_W + 2 * PADR)   // 68
#define LDS_H   (TILE_H + 2 * PADR)   // 68
#define LDS_N   (LDS_W * LDS_H)       // 4624 floats = 18.5 KB
#define NTHR    256                   // 8 wave32s

__global__ __launch_bounds__(NTHR)
void bilateral5x5_kernel(const float* __restrict__ x,
                         const float* __restrict__ sk,          // [25]
                         const float* __restrict__ sigma_color, // [1]
                         float* __restrict__ out) {
    __shared__ float tile[LDS_N];

    const int tid = threadIdx.x;
    const int x0  = blockIdx.x * TILE_W;
    const int y0  = blockIdx.y * TILE_H;
    const int img = blockIdx.z;                       // b*C + c
    const float* __restrict__ src = x + (size_t)img * IMG_H * IMG_W;

    // ---- Stage (TILE+4)^2 halo tile into LDS via CDNA5 async DMA ----
    // Reflect ('reflect' == no edge repeat) folded into per-lane global
    // addresses; async copy takes arbitrary per-lane global + LDS addresses,
    // so border handling costs nothing. Tracked by ASYNCcnt.
    for (int i = tid; i < LDS_N; i += NTHR) {
        const int ly = i / LDS_W;
        const int lx = i - ly * LDS_W;
        int gx = x0 + lx - PADR;
        int gy = y0 + ly - PADR;
        gx = (gx < 0) ? -gx : gx;
        gx = (gx > IMG_W - 1) ? (2 * (IMG_W - 1) - gx) : gx;
        gy = (gy < 0) ? -gy : gy;
        gy = (gy > IMG_H - 1) ? (2 * (IMG_H - 1) - gy) : gy;
        const float* gp = src + gy * IMG_W + gx;
        // Generic LDS pointer low 32 bits == wave-relative LDS byte address.
        unsigned lds_addr = (unsigned)(size_t)(&tile[i]);
        asm volatile("global_load_async_to_lds_b32 %0, %1, off"
                     :: "v"(lds_addr), "v"(gp)
                     : "memory");
    }
    asm volatile("s_wait_asynccnt 0" ::: "memory");
    __syncthreads();

    // ---- Weight math constants ----
    const float sc   = sigma_color[0];
    // exp(-d^2/(2 sc^2)) == exp2(d^2 * coef), coef = -1/(2 sc^2 ln2)
    const float coef = -1.4426950408889634f / (2.0f * sc * sc);

    // Spatial kernel: uniform address + constant index -> scalar loads.
    float skr[KS * KS];
#pragma unroll
    for (int t = 0; t < KS * KS; ++t) skr[t] = sk[t];

    const int tx = tid & (TILE_W - 1);  // 0..63
    const int ty = tid >> 6;            // 0..3

#pragma unroll
    for (int r = 0; r < TILE_H / 4; ++r) {
        const int oy = ty + r * 4;      // 0..63, 4-row interleave
        const float c = tile[(oy + PADR) * LDS_W + (tx + PADR)];
        float acc  = 0.0f;
        float wsum = 0.0f;
#pragma unroll
        for (int dy = 0; dy < KS; ++dy) {
#pragma unroll
            for (int dx = 0; dx < KS; ++dx) {
                const float p = tile[(oy + dy) * LDS_W + (tx + dx)];
                const float d = p - c;
                const float w = skr[dy * KS + dx] *
                                __builtin_amdgcn_exp2f(d * d * coef);
                wsum += w;
                acc = __builtin_fmaf(w, p, acc);
            }
        }
        const float inv = __builtin_amdgcn_rcpf(wsum + 1e-8f);
        out[(size_t)img * IMG_H * IMG_W + (size_t)(y0 + oy) * IMG_W + (x0 + tx)]
            = acc * inv;
    }
}

extern "C" void kernel_launch(void* const* d_in, const int* in_sizes, int n_in,
                              void* d_out, int out_size, void* d_ws, size_t ws_size,
                              hipStream_t stream) {
    const float* x   = (const float*)d_in[0];   // [4,3,512,512] f32
    const float* sk  = (const float*)d_in[1];   // [5,5] f32
    const float* sc  = (const float*)d_in[2];   // [1] f32
    float* out       = (float*)d_out;

    const int nimg = in_sizes[0] / (IMG_H * IMG_W);  // 12
    dim3 grid(IMG_W / TILE_W, IMG_H / TILE_H, nimg); // 8 x 8 x 12
    bilateral5x5_kernel<<<grid, NTHR, 0, stream>>>(x, sk, sc, out);
}